// EmbGCNEncoder_40192303956494
// MI455X (gfx1250) — compile-verified
//
#include <hip/hip_runtime.h>
#include <hip/hip_bf16.h>

// ---------------------------------------------------------------------------
// EmbGCNEncoder for MI455X (gfx1250, wave32).
// Bandwidth-bound SpMM (f32 atomics) + f32 WMMA GEMM (v_wmma_f32_16x16x4_f32).
// ---------------------------------------------------------------------------

typedef float v2f __attribute__((ext_vector_type(2)));
typedef float v8f __attribute__((ext_vector_type(8)));

#define GCN_D   128
#define LDA_PAD 132   // 16-row A tile stride in floats; 132%64=4 -> conflict-free

// ---- generic zero fill (used for degree counters and agg buffer) ----------
__global__ void zero_f32_kernel(float* __restrict__ p, long nelem) {
    long i = (long)blockIdx.x * blockDim.x + threadIdx.x;
    long stride = (long)gridDim.x * blockDim.x;
    for (; i < nelem; i += stride) p[i] = 0.0f;
}

// ---- degree counting (int atomics) ----------------------------------------
__global__ void degree_kernel(const int* __restrict__ src, const int* __restrict__ dst,
                              int* __restrict__ deg_out, int* __restrict__ deg_in, int e) {
    int i = blockIdx.x * blockDim.x + threadIdx.x;
    if (i < e) {
        atomicAdd(&deg_out[src[i]], 1);
        atomicAdd(&deg_in[dst[i]], 1);
    }
}

// ---- deg -> rsqrt norms ----------------------------------------------------
__global__ void norm_kernel(const int* __restrict__ deg_out, const int* __restrict__ deg_in,
                            float* __restrict__ norm_src, float* __restrict__ norm_dst, int n) {
    int i = blockIdx.x * blockDim.x + threadIdx.x;
    if (i < n) {
        int dov = deg_out[i]; if (dov < 1) dov = 1;
        int div = deg_in[i];  if (div < 1) div = 1;
        norm_src[i] = rsqrtf((float)dov);
        norm_dst[i] = rsqrtf((float)div);
    }
}

// ---- embedding lookup fused with norm_src pre-scale (wave per row) --------
__global__ __launch_bounds__(256)
void embed_scale_kernel(const int* __restrict__ batch, const float* __restrict__ emb,
                        const float* __restrict__ norm_src, float* __restrict__ hs, int n) {
    int row  = (blockIdx.x * 256 + threadIdx.x) >> 5;   // wave per node row
    int lane = threadIdx.x & 31;
    if (row >= n) return;
    int b    = batch[row];
    float ns = norm_src[row];
    float4 v = ((const float4*)(emb + (size_t)b * GCN_D))[lane];
    v.x *= ns; v.y *= ns; v.z *= ns; v.w *= ns;
    ((float4*)(hs + (size_t)row * GCN_D))[lane] = v;
}

// ---- edge scatter: agg[dst] += hs[src]  (wave per edge, hw f32 atomics) ---
__global__ __launch_bounds__(256)
void edge_scatter_kernel(const int* __restrict__ src, const int* __restrict__ dst,
                         const float* __restrict__ hs, float* __restrict__ agg, int e) {
    int edge = (blockIdx.x * 256 + threadIdx.x) >> 5;   // wave per edge
    int lane = threadIdx.x & 31;
    if (edge >= e) return;
    int s = src[edge];
    int d = dst[edge];
    float4 v = ((const float4*)(hs + (size_t)s * GCN_D))[lane];   // 512B/wave coalesced
    float* p = agg + (size_t)d * GCN_D + lane * 4;
    unsafeAtomicAdd(p + 0, v.x);   // global_atomic_add_f32
    unsafeAtomicAdd(p + 1, v.y);
    unsafeAtomicAdd(p + 2, v.z);
    unsafeAtomicAdd(p + 3, v.w);
}

// ---- GEMM: out = relu((agg*norm_dst) @ W + b) [* norm_src if !last] -------
// Block: 256 threads = 8 waves; one 16-row node stripe per block.
// Each wave computes one 16x16 output tile using v_wmma_f32_16x16x4_f32.
__global__ __launch_bounds__(256)
void gcn_gemm_wmma_kernel(const float* __restrict__ agg, const float* __restrict__ norm_dst,
                          const float* __restrict__ W, const float* __restrict__ bias,
                          const float* __restrict__ norm_src, float* __restrict__ out,
                          int n, int scale_out) {
    __shared__ float lW[64 * GCN_D];      // 32 KB : half of W (rows k..k+63)
    __shared__ float lA[16 * LDA_PAD];    // ~8.4 KB : scaled A tile

    const int tid     = threadIdx.x;
    const int rowBase = blockIdx.x * 16;

    // Stage A tile (16 x 128), scaled by norm_dst, padded stride.
    for (int t = tid; t < 512; t += 256) {            // 512 float4 chunks
        int r   = t >> 5;                             // 0..15 (row in tile)
        int c4  = t & 31;                             // 0..31 (float4 col)
        int row = rowBase + r;
        float4 v = make_float4(0.f, 0.f, 0.f, 0.f);
        float nd = 0.f;
        if (row < n) {
            v  = ((const float4*)(agg + (size_t)row * GCN_D))[c4];
            nd = norm_dst[row];
        }
        float* dst0 = &lA[r * LDA_PAD + c4 * 4];
        dst0[0] = v.x * nd; dst0[1] = v.y * nd; dst0[2] = v.z * nd; dst0[3] = v.w * nd;
    }

    const int lane = tid & 31;
    const int wave = tid >> 5;
    const int n16  = wave * 16;        // this wave's 16-col output tile
    const int mcol = lane & 15;        // A row / B,N column index
    const int hi   = lane >> 4;        // 0: K{0,1}/M0-7, 1: K{2,3}/M8-15
    const int koff = hi * 2;

    v8f c = {0.f, 0.f, 0.f, 0.f, 0.f, 0.f, 0.f, 0.f};

    #pragma unroll
    for (int p = 0; p < 2; ++p) {
        __syncthreads();               // A staged (p==0) / previous-phase lW readers done
        // Stage W rows [p*64, p*64+64) : 2048 float4
        {
            const float4* Wv  = (const float4*)(W + (size_t)p * 64 * GCN_D);
            float4*       lWv = (float4*)lW;
            #pragma unroll
            for (int i = 0; i < 8; ++i)
                lWv[tid + i * 256] = Wv[tid + i * 256];
        }
        __syncthreads();

        #pragma unroll
        for (int k = 0; k < 64; k += 4) {
            v2f a, b;
            // A fragment: 16x4 f32 — lane%16 = M, lane/16 picks K pair
            a[0] = lA[mcol * LDA_PAD + p * 64 + k + koff + 0];
            a[1] = lA[mcol * LDA_PAD + p * 64 + k + koff + 1];
            // B fragment: 4x16 f32 — lanes = N, half-wave picks K pair
            b[0] = lW[(k + koff + 0) * GCN_D + n16 + mcol];
            b[1] = lW[(k + koff + 1) * GCN_D + n16 + mcol];
            c = __builtin_amdgcn_wmma_f32_16x16x4_f32(
                    /*neg_a=*/false, a, /*neg_b=*/false, b,
                    /*c_mod=*/(short)0, c, /*reuse_a=*/false, /*reuse_b=*/false);
        }
    }

    // Epilogue: bias + ReLU (+ norm_src pre-scale for next layer's messages)
    const float bb = bias[n16 + mcol];
    #pragma unroll
    for (int v = 0; v < 8; ++v) {
        int m   = v + hi * 8;          // C/D layout: VGPR v, lane-half picks M / M+8
        int row = rowBase + m;
        if (row < n) {
            float val = c[v] + bb;
            val = val > 0.f ? val : 0.f;
            if (scale_out) val *= norm_src[row];
            out[(size_t)row * GCN_D + n16 + mcol] = val;
        }
    }
}

// ---------------------------------------------------------------------------
extern "C" void kernel_launch(void* const* d_in, const int* in_sizes, int n_in,
                              void* d_out, int out_size, void* d_ws, size_t ws_size,
                              hipStream_t stream) {
    const int*   batch = (const int*)d_in[0];
    const int*   src   = (const int*)d_in[1];
    const int*   dst   = (const int*)d_in[2];
    const float* emb   = (const float*)d_in[3];
    const float* Ws[3] = {(const float*)d_in[4], (const float*)d_in[6], (const float*)d_in[8]};
    const float* bs[3] = {(const float*)d_in[5], (const float*)d_in[7], (const float*)d_in[9]};

    const int n = in_sizes[0];     // nodes
    const int e = in_sizes[1];     // edges

    // Workspace carve-out (~52 MB for N=50K, D=128)
    char* w = (char*)d_ws;
    float* norm_src = (float*)w;  w += (size_t)n * sizeof(float);
    float* norm_dst = (float*)w;  w += (size_t)n * sizeof(float);
    int*   deg      = (int*)w;    w += (size_t)2 * n * sizeof(int);   // [deg_out | deg_in]
    float* hs       = (float*)w;  w += (size_t)n * GCN_D * sizeof(float);
    float* agg      = (float*)w;  w += (size_t)n * GCN_D * sizeof(float);
    float* out      = (float*)d_out;

    // 1) degrees -> norms
    zero_f32_kernel<<<1024, 256, 0, stream>>>((float*)deg, (long)2 * n);
    degree_kernel<<<(e + 255) / 256, 256, 0, stream>>>(src, dst, deg, deg + n, e);
    norm_kernel<<<(n + 255) / 256, 256, 0, stream>>>(deg, deg + n, norm_src, norm_dst, n);

    // 2) x = emb[batch] * norm_src  (pre-scaled messages)
    embed_scale_kernel<<<(n + 7) / 8, 256, 0, stream>>>(batch, emb, norm_src, hs, n);

    // 3) three GCN layers
    for (int l = 0; l < 3; ++l) {
        const int last = (l == 2);
        zero_f32_kernel<<<4096, 256, 0, stream>>>(agg, (long)n * GCN_D);
        edge_scatter_kernel<<<(e + 7) / 8, 256, 0, stream>>>(src, dst, hs, agg, e);
        gcn_gemm_wmma_kernel<<<(n + 15) / 16, 256, 0, stream>>>(
            agg, norm_dst, Ws[l], bs[l], norm_src, last ? out : hs, n, last ? 0 : 1);
    }
}